// BatchedEdges_32031866094387
// MI455X (gfx1250) — compile-verified
//
#include <hip/hip_runtime.h>

typedef float v2f __attribute__((ext_vector_type(2)));
typedef float v8f __attribute__((ext_vector_type(8)));

#define B_ 16
#define R_ 64
#define S_ 128
#define M_ 64
#define L_ 128
#define E_ 4032

// d_out layout (flat float offsets, reference return order)
#define OUT_INC_ADD   0
#define OUT_INC_GAIN  131072                    // B*R*L
#define OUT_SCAT_MEAN 262144                    // + B*R*L
#define OUT_SCAT_LOG  (262144 + 4194304)        // + B*R*R*M
#define OUT_SCAT_MSG  (262144 + 2*4194304)
#define OUT_TOTAL     12845056

__global__ void zero_out_kernel(float* __restrict__ out) {
  size_t i = (size_t)blockIdx.x * blockDim.x + threadIdx.x;
  float4 z = {0.f, 0.f, 0.f, 0.f};
  reinterpret_cast<float4*>(out)[i] = z;   // grid covers OUT_TOTAL/4 exactly
}

__launch_bounds__(128, 1)
__global__ void edge_kernel(const float* __restrict__ source,
                            const float* __restrict__ mean_w,
                            const float* __restrict__ mean_b,
                            const float* __restrict__ logstd_w,
                            const float* __restrict__ logstd_b,
                            const float* __restrict__ add_w,
                            const float* __restrict__ gain_w,
                            const int*   __restrict__ src_idx,
                            const int*   __restrict__ tgt_idx,
                            float* __restrict__ out)
{
  __shared__ float s_src[B_][S_ + 4];   // 16 x 132, pad kills bank conflicts
  __shared__ float s_msg[B_][M_ + 4];   // 16 x 68

  const int e    = blockIdx.x;
  const int tid  = threadIdx.x;
  const int lane = tid & 31;
  const int w    = tid >> 5;            // wave id 0..3
  const int hi   = lane >> 4;           // upper half-wave holds K+2
  const int ln   = lane & 15;

  const int s_row = src_idx[e];
  const int t_row = tgt_idx[e];

  // ---- stage 0: gather src tile (b, src_idx[e], :) -> LDS --------------
  for (int i = tid; i < B_ * S_; i += 128) {
    int b = i >> 7, c = i & 127;
    s_src[b][c] = source[((size_t)b * R_ + s_row) * S_ + c];
  }
  __syncthreads();

  // ---- stage 1: mean & logstd (16x64 each), wave w -> m-cols [16w,16w+16)
  // WMMA f32 16x16x4: A lane = {row ln, K = k0+2*hi .. +1}; B lane = {col ln}
  v8f cMean = {};
  v8f cLog  = {};
  const float* wm   = mean_w   + ((size_t)e * M_ + 16*w + ln) * S_ + 2*hi;
  const float* wl   = logstd_w + ((size_t)e * M_ + 16*w + ln) * S_ + 2*hi;
  const float* arow = &s_src[ln][2*hi];
  #pragma unroll 8
  for (int k0 = 0; k0 < S_; k0 += 4) {
    v2f a  = { arow[k0], arow[k0 + 1] };
    v2f bm = { wm[k0],   wm[k0 + 1] };
    v2f bl = { wl[k0],   wl[k0 + 1] };
    cMean = __builtin_amdgcn_wmma_f32_16x16x4_f32(false, a, false, bm,
                                                  (short)0, cMean, false, false);
    cLog  = __builtin_amdgcn_wmma_f32_16x16x4_f32(false, a, false, bl,
                                                  (short)0, cLog,  false, false);
  }

  // bias is per output column m = 16w+ln (same for all 8 C rows in a lane)
  const float biasM = mean_b  [(size_t)e * M_ + 16*w + ln];
  const float biasL = logstd_b[(size_t)e * M_ + 16*w + ln];

  // scat stores: idx = ((b*R + s)*R + t)*M + m
  const size_t scat_off = (((size_t)s_row * R_) + t_row) * M_ + 16*w + ln;
  float* o_mean = out + OUT_SCAT_MEAN;
  float* o_log  = out + OUT_SCAT_LOG;
  float* o_msg  = out + OUT_SCAT_MSG;
  #pragma unroll
  for (int i = 0; i < 8; ++i) {
    float mv = cMean[i] + biasM;
    float lv = cLog[i]  + biasL;
    int b = i + 8*hi;                          // C/D row layout: vgpr i -> M=i / M=i+8
    s_msg[b][16*w + ln] = mv;
    size_t idx = (size_t)b * (R_ * R_ * M_) + scat_off;
    o_mean[idx] = mv;
    o_msg [idx] = mv;                          // msg == mean (deterministic path)
    o_log [idx] = lv;
  }
  __syncthreads();

  // ---- stage 2: add & gain (16x128 each); wave w -> l-groups {w, w+4} ---
  v8f cA0 = {}, cA1 = {}, cG0 = {}, cG1 = {};
  const int g0 = w, g1 = w + 4;
  const float* wa0 = add_w  + ((size_t)e * L_ + 16*g0 + ln) * M_ + 2*hi;
  const float* wa1 = add_w  + ((size_t)e * L_ + 16*g1 + ln) * M_ + 2*hi;
  const float* wg0 = gain_w + ((size_t)e * L_ + 16*g0 + ln) * M_ + 2*hi;
  const float* wg1 = gain_w + ((size_t)e * L_ + 16*g1 + ln) * M_ + 2*hi;
  const float* mrow = &s_msg[ln][2*hi];
  #pragma unroll 4
  for (int k0 = 0; k0 < M_; k0 += 4) {
    v2f a  = { mrow[k0], mrow[k0 + 1] };
    v2f b0 = { wa0[k0],  wa0[k0 + 1] };
    v2f b1 = { wa1[k0],  wa1[k0 + 1] };
    v2f b2 = { wg0[k0],  wg0[k0 + 1] };
    v2f b3 = { wg1[k0],  wg1[k0 + 1] };
    cA0 = __builtin_amdgcn_wmma_f32_16x16x4_f32(false, a, false, b0,
                                                (short)0, cA0, false, false);
    cA1 = __builtin_amdgcn_wmma_f32_16x16x4_f32(false, a, false, b1,
                                                (short)0, cA1, false, false);
    cG0 = __builtin_amdgcn_wmma_f32_16x16x4_f32(false, a, false, b2,
                                                (short)0, cG0, false, false);
    cG1 = __builtin_amdgcn_wmma_f32_16x16x4_f32(false, a, false, b3,
                                                (short)0, cG1, false, false);
  }

  // scatter-add into inc_add / inc_gain at target row t
  float* o_add  = out + OUT_INC_ADD;
  float* o_gain = out + OUT_INC_GAIN;
  #pragma unroll
  for (int i = 0; i < 8; ++i) {
    int b = i + 8*hi;
    size_t base = ((size_t)b * R_ + t_row) * L_;
    unsafeAtomicAdd(&o_add [base + 16*g0 + ln], cA0[i]);
    unsafeAtomicAdd(&o_add [base + 16*g1 + ln], cA1[i]);
    unsafeAtomicAdd(&o_gain[base + 16*g0 + ln], cG0[i]);
    unsafeAtomicAdd(&o_gain[base + 16*g1 + ln], cG1[i]);
  }
}

extern "C" void kernel_launch(void* const* d_in, const int* in_sizes, int n_in,
                              void* d_out, int out_size, void* d_ws, size_t ws_size,
                              hipStream_t stream) {
  const float* source   = (const float*)d_in[0];
  const float* mean_w   = (const float*)d_in[1];
  const float* mean_b   = (const float*)d_in[2];
  const float* logstd_w = (const float*)d_in[3];
  const float* logstd_b = (const float*)d_in[4];
  const float* add_w    = (const float*)d_in[5];
  const float* gain_w   = (const float*)d_in[6];
  const int*   src_idx  = (const int*)d_in[7];
  const int*   tgt_idx  = (const int*)d_in[8];
  float* out = (float*)d_out;

  // zero outputs (atomics accumulate on top; scat diagonal must stay zero)
  zero_out_kernel<<<(OUT_TOTAL / 4) / 256, 256, 0, stream>>>(out);

  // one workgroup (4 wave32) per edge: streams each weight byte exactly once
  edge_kernel<<<E_, 128, 0, stream>>>(source, mean_w, mean_b, logstd_w, logstd_b,
                                      add_w, gain_w, src_idx, tgt_idx, out);
}